// TwoRDM_79001628443210
// MI455X (gfx1250) — compile-verified
//
#include <hip/hip_runtime.h>
#include <hip/hip_bf16.h>
#include <cstdint>

#define NS   48
#define N2   2304                      // NS*NS
#define NN   ((size_t)N2 * (size_t)N2) // 5,308,416 elements per matrix

#define BM 128
#define BN 64
#define BK 32
#define LDA 36                         // padded A row stride (dwords); 16 rows -> 16 banks
#define LDBP 160                       // interleaved-B row stride (dwords); halves -> disjoint banks
#define NS_ITERS 22
#define NPART 2592                     // NN / 2048

typedef __attribute__((ext_vector_type(2))) float v2f;
typedef __attribute__((ext_vector_type(8))) float v8f;
typedef __attribute__((__vector_size__(4 * sizeof(int)))) int v4i_t;

#if defined(__AMDGCN__) && \
    __has_builtin(__builtin_amdgcn_global_load_async_to_lds_b128) && \
    __has_builtin(__builtin_amdgcn_global_load_async_to_lds_b32)
#define USE_ASYNC_LDS 1
#else
#define USE_ASYNC_LDS 0
#endif

__device__ __forceinline__ void wait_async_lds()
{
#if defined(__AMDGCN__)
#if __has_builtin(__builtin_amdgcn_s_wait_asynccnt)
    __builtin_amdgcn_s_wait_asynccnt(0);
#else
    asm volatile("s_wait_asynccnt 0x0" ::: "memory");
#endif
#endif
}

// ---------------------------------------------------------------------------
// D = alpha * (A @ B) + beta * C     (all N2 x N2, row-major, f32)
// V_WMMA_F32_16X16X4_F32; 8 wave32 / block; block tile 128x64, 32x32 per wave.
// Double-buffered LDS; async global->LDS staging (ASYNCcnt) when available.
// B is stored k-pair interleaved so each B fragment is one ds_load_b64.
// ---------------------------------------------------------------------------
__global__ __launch_bounds__(256, 2)
void gemm_f32_wmma(const float* __restrict__ A, const float* __restrict__ B,
                   const float* C, float* __restrict__ D,
                   float alpha, float beta)
{
    __shared__ float As[2][BM][LDA];        // 2 * 128*36*4 = 36 KB
    __shared__ float Bp[2][BK / 2][LDBP];   // 2 * 16*160*4 = 20 KB

    const int tid  = threadIdx.x;
    const int wave = tid >> 5;
    const int lane = tid & 31;
    const int half = lane >> 4;   // 0: K even pair / M rows 0..7 ; 1: K odd pair / rows 8..15
    const int lrow = lane & 15;

    const int blockM = blockIdx.y * BM;
    const int blockN = blockIdx.x * BN;
    const int waveM  = (wave >> 1) * 32;
    const int waveN  = (wave & 1) * 32;

    v8f acc00 = {}, acc01 = {}, acc10 = {}, acc11 = {};

#if USE_ASYNC_LDS
    // ---- async staging (per-wave issues, tracked by ASYNCcnt) ----
#define ISSUE_TILE(KB, BUF)                                                              \
    {                                                                                    \
        _Pragma("unroll")                                                                \
        for (int j = 0; j < 4; j++) { /* A: 1024 float4 chunks, b128 per lane */         \
            int f = tid + j * 256;                                                       \
            int row = f >> 3, c4 = (f & 7) << 2;                                         \
            const float* g = A + (size_t)(blockM + row) * N2 + ((KB) + c4);              \
            __builtin_amdgcn_global_load_async_to_lds_b128(                              \
                (v4i_t*)g, (v4i_t*)&As[BUF][row][c4], 0, 0);                             \
        }                                                                                \
        _Pragma("unroll")                                                                \
        for (int j = 0; j < 8; j++) { /* B: 2048 elems, interleaved scatter, b32 */      \
            int e = tid + j * 256;                                                       \
            int k = e >> 6, col = e & 63;                                                \
            const float* g = B + (size_t)((KB) + k) * N2 + (blockN + col);               \
            __builtin_amdgcn_global_load_async_to_lds_b32(                               \
                (int*)g, (int*)&Bp[BUF][k >> 1][col * 2 + (k & 1)], 0, 0);               \
        }                                                                                \
    }

    ISSUE_TILE(0, 0)
    wait_async_lds();
    __syncthreads();
#else
    // ---- register-pipelined fallback ----
    float4 ra[4];
    float4 rb[2];
#define LOAD_REGS(KB)                                                                    \
    {                                                                                    \
        _Pragma("unroll")                                                                \
        for (int j = 0; j < 4; j++) {                                                    \
            int f = tid + j * 256;                                                       \
            int row = f >> 3, c4 = (f & 7) << 2;                                         \
            ra[j] = *(const float4*)(A + (size_t)(blockM + row) * N2 + ((KB) + c4));     \
        }                                                                                \
        _Pragma("unroll")                                                                \
        for (int j = 0; j < 2; j++) {                                                    \
            int f = tid + j * 256;                                                       \
            int k = f >> 4, c4 = (f & 15) << 2;                                          \
            rb[j] = *(const float4*)(B + (size_t)((KB) + k) * N2 + (blockN + c4));       \
        }                                                                                \
    }
#define STORE_LDS(BUF)                                                                   \
    {                                                                                    \
        _Pragma("unroll")                                                                \
        for (int j = 0; j < 4; j++) {                                                    \
            int f = tid + j * 256;                                                       \
            int row = f >> 3, c4 = (f & 7) << 2;                                         \
            *(float4*)(&As[BUF][row][c4]) = ra[j];                                       \
        }                                                                                \
        _Pragma("unroll")                                                                \
        for (int j = 0; j < 2; j++) {                                                    \
            int f = tid + j * 256;                                                       \
            int k = f >> 4, c4 = (f & 15) << 2;                                          \
            int par = k & 1;                                                             \
            float* dst = &Bp[BUF][k >> 1][0];                                            \
            dst[(c4 + 0) * 2 + par] = rb[j].x;                                           \
            dst[(c4 + 1) * 2 + par] = rb[j].y;                                           \
            dst[(c4 + 2) * 2 + par] = rb[j].z;                                           \
            dst[(c4 + 3) * 2 + par] = rb[j].w;                                           \
        }                                                                                \
    }

    LOAD_REGS(0)
    STORE_LDS(0)
    __syncthreads();
#endif

    int p = 0;
    for (int kb = 0; kb < N2; kb += BK) {
        const bool has_next = (kb + BK) < N2;   // uniform

#if USE_ASYNC_LDS
        if (has_next) ISSUE_TILE(kb + BK, p ^ 1)
#else
        if (has_next) LOAD_REGS(kb + BK)
#endif

        const float (*Ac)[LDA]  = As[p];
        const float (*Bc)[LDBP] = Bp[p];
#pragma unroll
        for (int kk = 0; kk < BK; kk += 4) {
            // A fragment: lanes 0-15 K=kk..kk+1, lanes 16-31 K=kk+2..kk+3 (one b64 each)
            const int kA = kk + 2 * half;
            v2f a0 = *(const v2f*)&Ac[waveM + lrow][kA];
            v2f a1 = *(const v2f*)&Ac[waveM + 16 + lrow][kA];
            // B fragment: interleaved pair (kA, kA+1) contiguous -> one b64 each
            const int kp = (kk >> 1) + half;
            v2f b0 = *(const v2f*)&Bc[kp][(waveN + lrow) * 2];
            v2f b1 = *(const v2f*)&Bc[kp][(waveN + 16 + lrow) * 2];

            acc00 = __builtin_amdgcn_wmma_f32_16x16x4_f32(false, a0, false, b0, (short)0, acc00, false, false);
            acc01 = __builtin_amdgcn_wmma_f32_16x16x4_f32(false, a0, false, b1, (short)0, acc01, false, false);
            acc10 = __builtin_amdgcn_wmma_f32_16x16x4_f32(false, a1, false, b0, (short)0, acc10, false, false);
            acc11 = __builtin_amdgcn_wmma_f32_16x16x4_f32(false, a1, false, b1, (short)0, acc11, false, false);
        }

#if USE_ASYNC_LDS
        wait_async_lds();
#else
        if (has_next) STORE_LDS(p ^ 1)
#endif
        __syncthreads();
        p ^= 1;
    }

    // Epilogue: C/D layout -> vgpr v holds row (v + 8*half), col = lane&15
#define STORE_TILE(ACC, MT, NT)                                                          \
    {                                                                                    \
        _Pragma("unroll")                                                                \
        for (int v = 0; v < 8; v++) {                                                    \
            size_t idx = (size_t)(blockM + waveM + (MT) + v + 8 * half) * N2             \
                       + (size_t)(blockN + waveN + (NT) + lrow);                         \
            float r = alpha * ACC[v];                                                    \
            if (beta != 0.0f) r += beta * C[idx];                                        \
            D[idx] = r;                                                                  \
        }                                                                                \
    }
    STORE_TILE(acc00, 0, 0)
    STORE_TILE(acc01, 0, 16)
    STORE_TILE(acc10, 16, 0)
    STORE_TILE(acc11, 16, 16)
#undef STORE_TILE
}

// ---------------------------------------------------------------------------
// out = 0.5 * (in + in^T)  (matrix hermitize of the flattened 2304x2304)
// ---------------------------------------------------------------------------
__global__ void k_hermitize(const float* __restrict__ in, float* __restrict__ out)
{
    int r = blockIdx.y;
    int c = blockIdx.x * 256 + threadIdx.x;
    out[(size_t)r * N2 + c] = 0.5f * (in[(size_t)r * N2 + c] + in[(size_t)c * N2 + r]);
}

// ---------------------------------------------------------------------------
// Deterministic two-pass Frobenius-norm reduction
// ---------------------------------------------------------------------------
__global__ void k_frob1(const float* __restrict__ M, float* __restrict__ partials)
{
    __shared__ float sm[256];
    int tid = threadIdx.x;
    size_t base = (size_t)blockIdx.x * 2048 + tid;
    float s = 0.f;
#pragma unroll
    for (int j = 0; j < 8; j++) { float v = M[base + j * 256]; s += v * v; }
    sm[tid] = s; __syncthreads();
    for (int st = 128; st > 0; st >>= 1) {
        if (tid < st) sm[tid] += sm[tid + st];
        __syncthreads();
    }
    if (tid == 0) partials[blockIdx.x] = sm[0];
}

__global__ void k_frob2(const float* __restrict__ partials, float* __restrict__ scal)
{
    __shared__ float sm[256];
    int tid = threadIdx.x;
    float s = 0.f;
    for (int i = tid; i < NPART; i += 256) s += partials[i];
    sm[tid] = s; __syncthreads();
    for (int st = 128; st > 0; st >>= 1) {
        if (tid < st) sm[tid] += sm[tid + st];
        __syncthreads();
    }
    if (tid == 0) { scal[0] = sm[0]; scal[1] = rsqrtf(fmaxf(sm[0], 1e-30f)); }
}

__global__ void k_scale(const float* __restrict__ in, float* __restrict__ out,
                        const float* __restrict__ sptr)
{
    size_t i = (size_t)blockIdx.x * 256 + threadIdx.x;
    out[i] = in[i] * sptr[0];
}

// ---------------------------------------------------------------------------
// In-place symmetrize under (i,k,j,l) -> (k,i,l,j): pairs (r,c)<->(P r, P c)
// Each unordered pair handled by exactly one thread -> race-free.
// ---------------------------------------------------------------------------
__global__ void k_symmetrize_pair(float* __restrict__ M)
{
    size_t idx = (size_t)blockIdx.x * 256 + threadIdx.x;
    int r = (int)(idx / N2), c = (int)(idx % N2);
    int pr = (r % NS) * NS + (r / NS);
    int pc = (c % NS) * NS + (c / NS);
    size_t idx2 = (size_t)pr * N2 + pc;
    if (idx <= idx2) {
        float v = 0.5f * (M[idx] + M[idx2]);
        M[idx] = v;
        M[idx2] = v;
    }
}

// ---------------------------------------------------------------------------
// Reductions: trace + ring-hopping sum (for uu / dd blocks)
// ---------------------------------------------------------------------------
__global__ void k_reduce_ud(const float* __restrict__ M,
                            float* __restrict__ trOut, float* __restrict__ kinOut)
{
    __shared__ float s1[256], s2[256];
    int tid = threadIdx.x;
    float tr = 0.f, kin = 0.f;
    for (int r = tid; r < N2; r += 256) tr += M[(size_t)r * N2 + r];
    for (int p = tid; p < NS * NS; p += 256) {
        int i = p / NS, k = p % NS;
        size_t row = (size_t)(i * NS + k) * N2;
        int ip = (i + 1) % NS, im = (i + NS - 1) % NS;
        kin += M[row + ip * NS + k] + M[row + im * NS + k];
    }
    s1[tid] = tr; s2[tid] = kin; __syncthreads();
    for (int st = 128; st > 0; st >>= 1) {
        if (tid < st) { s1[tid] += s1[tid + st]; s2[tid] += s2[tid + st]; }
        __syncthreads();
    }
    if (tid == 0) { trOut[0] = s1[0]; kinOut[0] = s2[0]; }
}

// trace, kin(01 pattern), kin(10 pattern = P.P), sum_{ijji}, sum_{iiii}
__global__ void k_reduce_du(const float* __restrict__ M, float* __restrict__ out5)
{
    __shared__ float sm[5][256];
    int tid = threadIdx.x;
    float tr = 0.f, k01 = 0.f, k10 = 0.f, sij = 0.f, sii = 0.f;
    for (int r = tid; r < N2; r += 256) {
        tr += M[(size_t)r * N2 + r];
        int pr = (r % NS) * NS + (r / NS);
        sij += M[(size_t)r * N2 + pr];
    }
    for (int p = tid; p < NS * NS; p += 256) {
        int i = p / NS, k = p % NS;
        int ip = (i + 1) % NS, im = (i + NS - 1) % NS;
        k01 += M[(size_t)(i * NS + k) * N2 + ip * NS + k]
             + M[(size_t)(i * NS + k) * N2 + im * NS + k];
        k10 += M[(size_t)(k * NS + i) * N2 + k * NS + ip]
             + M[(size_t)(k * NS + i) * N2 + k * NS + im];
    }
    for (int i = tid; i < NS; i += 256) {
        int r = i * NS + i;
        sii += M[(size_t)r * N2 + r];
    }
    sm[0][tid] = tr; sm[1][tid] = k01; sm[2][tid] = k10; sm[3][tid] = sij; sm[4][tid] = sii;
    __syncthreads();
    for (int st = 128; st > 0; st >>= 1) {
        if (tid < st)
            for (int q = 0; q < 5; q++) sm[q][tid] += sm[q][tid + st];
        __syncthreads();
    }
    if (tid == 0)
        for (int q = 0; q < 5; q++) out5[q] = sm[q][0];
}

// ---------------------------------------------------------------------------
// Assemble (E, loss6, loss7). Trace targets: diag 24*23=552, off-diag 24*24=576.
// ---------------------------------------------------------------------------
__global__ void k_finalize(const float* __restrict__ scal, const float* __restrict__ tp,
                           const float* __restrict__ Up, float* __restrict__ out)
{
    float trDD = scal[2], kinDD = scal[3];
    float trUU = scal[4], kinUU = scal[5];
    float trDU = scal[6], k01 = scal[7], k10 = scal[8], sij = scal[9], sii = scal[10];

    float n00 = 552.0f / trDD, n11 = 552.0f / trUU;
    float n01 = 576.0f / trDU, n10 = 576.0f / trDU;

    float d00 = 552.0f - n00 * trDD;
    float d01 = 576.0f - n01 * trDU;
    float d10 = 576.0f - n10 * trDU;
    float d11 = 552.0f - n11 * trUU;
    float loss6 = d00 * d00 + d01 * d01 + d10 * d10 + d11 * d11;

    float c7 = 24.0f + n01 * sij;       // Sz = 0
    float loss7 = c7 * c7;              // target S(S+1) = 0

    float t = tp[0], U = Up[0];
    float Ekin = -t * (n00 * kinDD + n01 * k01 + n10 * k10 + n11 * kinUU) / 47.0f;
    float EU = U * n01 * sii;

    out[0] = Ekin + EU;
    out[1] = loss6;
    out[2] = loss7;
}

// ---------------------------------------------------------------------------
extern "C" void kernel_launch(void* const* d_in, const int* in_sizes, int n_in,
                              void* d_out, int out_size, void* d_ws, size_t ws_size,
                              hipStream_t stream)
{
    (void)in_sizes; (void)n_in; (void)out_size; (void)ws_size;
    const float* upup = (const float*)d_in[0];
    const float* dndn = (const float*)d_in[1];
    const float* dnup = (const float*)d_in[2];
    const float* tp   = (const float*)d_in[3];
    const float* Up   = (const float*)d_in[4];
    float* out = (float*)d_out;

    float* scal     = (float*)d_ws;     // [0..63] scalar slots
    float* partials = scal + 64;        // [64..64+NPART) frob partials
    float* big      = scal + 8192;      // 4 big 2304^2 buffers (aligned)
    float* bM = big;                    // hermitized input (kept for final step)
    float* b0 = big + NN;
    float* b1 = big + 2 * NN;
    float* b2 = big + 3 * NN;

    dim3 gGemm(N2 / BN, N2 / BM);       // (36, 18)
    dim3 gHerm(N2 / 256, N2);           // (9, 2304)
    int  gFlat = (int)(NN / 256);       // 20736

    // PSD projection via Newton-Schulz sign iteration (all GEMMs on WMMA).
    // Result left in b1; b0/b2 ping-pong, bM preserved.
    auto run_psd = [&](const float* in) -> float* {
        k_hermitize<<<gHerm, 256, 0, stream>>>(in, bM);
        k_frob1<<<NPART, 256, 0, stream>>>(bM, partials);
        k_frob2<<<1, 256, 0, stream>>>(partials, scal);
        float* X = b0; float* T = b1; float* Y = b2;
        k_scale<<<gFlat, 256, 0, stream>>>(bM, X, scal + 1);   // X0 = M / ||M||_F
        for (int it = 0; it < NS_ITERS; ++it) {
            gemm_f32_wmma<<<gGemm, 256, 0, stream>>>(X, X, X, T, 1.0f,  0.0f); // T = X^2
            gemm_f32_wmma<<<gGemm, 256, 0, stream>>>(X, T, X, Y, -0.5f, 1.5f); // Y = 1.5X - 0.5 X T
            float* tmp = X; X = Y; Y = tmp;
        }
        // PSD(M) = 0.5*(M + M * sign(M))
        gemm_f32_wmma<<<gGemm, 256, 0, stream>>>(bM, X, bM, T, 0.5f, 0.5f);
        return T;  // always b1
    };

    float* R;
    R = run_psd(dndn);                                            // down-down
    k_reduce_ud<<<1, 256, 0, stream>>>(R, scal + 2, scal + 3);
    R = run_psd(upup);                                            // up-up
    k_reduce_ud<<<1, 256, 0, stream>>>(R, scal + 4, scal + 5);
    R = run_psd(dnup);                                            // down-up
    k_symmetrize_pair<<<gFlat, 256, 0, stream>>>(R);              // 0.5*(du + du^{(1,0,3,2)})
    k_reduce_du<<<1, 256, 0, stream>>>(R, scal + 6);

    k_finalize<<<1, 1, 0, stream>>>(scal, tp, Up, out);
}